// BiLSTM_59433757442055
// MI455X (gfx1250) — compile-verified
//
#include <hip/hip_runtime.h>
#include <math.h>

// ---------------------------------------------------------------------------
// Problem constants (from reference)
// ---------------------------------------------------------------------------
#define SEQ     512
#define D_W     100
#define D_P     25
#define D_E     300
#define D_IN    425          // 100 + 25 + 300
#define VEC_LD  428          // D_IN padded (zeros) to multiple of 4
#define HH      100          // LSTM hidden
#define G4H     400          // 4*H gate width
#define CAT_K   625          // D_IN + 2H
#define CAT_LD  628          // padded to multiple of 4
#define D_MLP   400
#define N_REL   42
#define NREL_P  48           // N_REL padded to WMMA tile multiple
#define G_LD    48

typedef float v2f __attribute__((ext_vector_type(2)));
typedef float v8f __attribute__((ext_vector_type(8)));

// ---------------------------------------------------------------------------
// 0) Pack a row-major weight matrix W[N][K] into Wp[Npad][Kpad], zero padded.
//    Removes ALL bounds checks from the GEMM hot loop.
// ---------------------------------------------------------------------------
__global__ void pack_w_kernel(const float* __restrict__ W, int N, int K,
                              float* __restrict__ Wp, int Kpad) {
    const int n = blockIdx.x;                       // 0 .. Npad-1
    float* dst = Wp + (long)n * Kpad;
    if (n < N) {
        const float* src = W + (long)n * K;
        for (int k = threadIdx.x; k < Kpad; k += blockDim.x)
            dst[k] = (k < K) ? src[k] : 0.0f;
    } else {
        for (int k = threadIdx.x; k < Kpad; k += blockDim.x)
            dst[k] = 0.0f;
    }
}

// ---------------------------------------------------------------------------
// 1) Gather embeddings -> vec[SEQ][VEC_LD], zero-padded cols [425,428)
// ---------------------------------------------------------------------------
__global__ void gather_vec_kernel(const int* __restrict__ widx,
                                  const int* __restrict__ pidx,
                                  const int* __restrict__ eidx,
                                  const float* __restrict__ wemb,
                                  const float* __restrict__ pemb,
                                  const float* __restrict__ eemb,
                                  float* __restrict__ vec) {
    const int t = blockIdx.x;
    const float* wr = wemb + (long)widx[t] * D_W;
    const float* pr = pemb + (long)pidx[t] * D_P;
    const float* er = eemb + (long)eidx[t] * D_E;
    float* out = vec + (long)t * VEC_LD;
    for (int c = threadIdx.x; c < VEC_LD; c += blockDim.x) {
        float v = 0.0f;
        if (c < D_W)             v = wr[c];
        else if (c < D_W + D_P)  v = pr[c - D_W];
        else if (c < D_IN)       v = er[c - (D_W + D_P)];
        out[c] = v;
    }
}

// ---------------------------------------------------------------------------
// 2) WMMA fp32 GEMM on pre-padded operands:
//      C[M, ldc] = A[M, Kpad] * Wp[Npad, Kpad]^T  (+ bscale*bias, col < Nbias)
//    One wave32 per 16x16 C tile; K stepped by 4 via V_WMMA_F32_16X16X4_F32,
//    unrolled x2 with two accumulators to break the D->C dependency chain.
//    Hot loop: 2x global_load_b64 (A) + 2x global_load_b64 (B) + 2x wmma.
//    ISA layouts (CDNA5 ISA 7.12.2):
//      A 16x4 : lanes 0-15 = rows M, half-wave (lane>=16) takes K pair {2,3}
//      B 4x16 : lanes 0-15 = cols N, half-wave takes K pair {2,3}
//      C 16x16: VGPR r holds row r (lanes 0-15) and row r+8 (lanes 16-31)
// ---------------------------------------------------------------------------
__global__ void wmma_gemm_f32(const float* __restrict__ A,
                              const float* __restrict__ Wp, int Kpad,
                              const float* __restrict__ bias, float bscale,
                              int Nbias,
                              float* __restrict__ C, int ldc, int tiles_n) {
    const int tile = blockIdx.x * (blockDim.x >> 5) + (threadIdx.x >> 5);
    const int lane = threadIdx.x & 31;
    const int mt = tile / tiles_n;
    const int nt = tile - mt * tiles_n;
    const int m0 = mt * 16;
    const int n0 = nt * 16;
    const int half = lane >> 4;      // 0: K{0,1}, 1: K{2,3}
    const int l    = lane & 15;      // row (A) / col (B) index within tile

    const float* arow = A  + (long)(m0 + l) * Kpad + half * 2;
    const float* wrow = Wp + (long)(n0 + l) * Kpad + half * 2;

    v8f acc0 = {};
    v8f acc1 = {};
    const int K8 = Kpad & ~7;
    int k = 0;
    for (; k < K8; k += 8) {
        v2f a0 = *(const v2f*)(arow + k);
        v2f b0 = *(const v2f*)(wrow + k);
        v2f a1 = *(const v2f*)(arow + k + 4);
        v2f b1 = *(const v2f*)(wrow + k + 4);
        acc0 = __builtin_amdgcn_wmma_f32_16x16x4_f32(
                   false, a0, false, b0, (short)0, acc0, false, false);
        acc1 = __builtin_amdgcn_wmma_f32_16x16x4_f32(
                   false, a1, false, b1, (short)0, acc1, false, false);
    }
    for (; k < Kpad; k += 4) {       // Kpad % 8 == 4 tail (428, 628)
        v2f a0 = *(const v2f*)(arow + k);
        v2f b0 = *(const v2f*)(wrow + k);
        acc0 = __builtin_amdgcn_wmma_f32_16x16x4_f32(
                   false, a0, false, b0, (short)0, acc0, false, false);
    }

    const int col = n0 + l;
    const float bv = (col < Nbias) ? bias[col] * bscale : 0.0f;
#pragma unroll
    for (int r = 0; r < 8; ++r) {
        C[(long)(m0 + r + half * 8) * ldc + col] = acc0[r] + acc1[r] + bv;
    }
}

// ---------------------------------------------------------------------------
// 3) LSTM cell epilogue (h0=c0=0, torch gate order i,f,g,o) + build
//    cat[SEQ][CAT_LD]:  t<=255 -> [h(200), vec(425), 0..], else [vec, h, 0..]
// ---------------------------------------------------------------------------
__device__ __forceinline__ float sigm(float x) { return 1.0f / (1.0f + expf(-x)); }

__global__ void lstm_cat_kernel(const float* __restrict__ gf,
                                const float* __restrict__ gb,
                                const float* __restrict__ vec,
                                float* __restrict__ cat) {
    __shared__ float sh[2 * HH];
    const int t   = blockIdx.x;
    const int tid = threadIdx.x;

    if (tid < 2 * HH) {
        const float* g = (tid < HH) ? (gf + (long)t * G4H) : (gb + (long)t * G4H);
        const int j = (tid < HH) ? tid : tid - HH;
        const float ii = g[j];              // input gate
        const float gg = g[2 * HH + j];     // cell candidate
        const float oo = g[3 * HH + j];     // output gate
        const float c  = sigm(ii) * tanhf(gg);   // f * c0 == 0
        sh[tid] = sigm(oo) * tanhf(c);
    }
    __syncthreads();

    const float* vrow = vec + (long)t * VEC_LD;
    float* crow = cat + (long)t * CAT_LD;
    const bool first_half = (t <= (SEQ - 1) / 2);
    for (int c = tid; c < CAT_LD; c += blockDim.x) {
        float v = 0.0f;
        if (first_half) {
            if (c < 2 * HH)       v = sh[c];
            else if (c < CAT_K)   v = vrow[c - 2 * HH];
        } else {
            if (c < D_IN)         v = vrow[c];
            else if (c < CAT_K)   v = sh[c - D_IN];
        }
        crow[c] = v;
    }
}

// ---------------------------------------------------------------------------
// 4) scores[i,j,k] = tanh(g[i,k] + g[j,k])   (0.5*b folded into g twice)
//    44 MB of fully coalesced stores; g (96 KB) stays L2-resident.
// ---------------------------------------------------------------------------
__global__ void scores_kernel(const float* __restrict__ g, float* __restrict__ out) {
    __shared__ float gi[N_REL];
    const int i = blockIdx.x;
    if (threadIdx.x < N_REL) gi[threadIdx.x] = g[(long)i * G_LD + threadIdx.x];
    __syncthreads();

    const int per_i = SEQ * N_REL;                 // 21504
    float* orow = out + (long)i * per_i;
    for (int e = threadIdx.x; e < per_i; e += blockDim.x) {
        const int j = e / N_REL;
        const int k = e - j * N_REL;
        orow[e] = tanhf(gi[k] + g[(long)j * G_LD + k]);
    }
}

// ---------------------------------------------------------------------------
// Host launch
// ---------------------------------------------------------------------------
extern "C" void kernel_launch(void* const* d_in, const int* in_sizes, int n_in,
                              void* d_out, int out_size, void* d_ws, size_t ws_size,
                              hipStream_t stream) {
    (void)in_sizes; (void)n_in; (void)out_size; (void)ws_size;

    const int*   word_idx  = (const int*)  d_in[0];
    const int*   pos_idx   = (const int*)  d_in[1];
    const int*   ext_idx   = (const int*)  d_in[2];
    const float* word_emb  = (const float*)d_in[3];
    const float* pos_emb   = (const float*)d_in[4];
    const float* ext_emb   = (const float*)d_in[5];
    const float* w_ih_f    = (const float*)d_in[6];
    const float* b_f       = (const float*)d_in[7];
    const float* w_ih_b    = (const float*)d_in[8];
    const float* b_b       = (const float*)d_in[9];
    const float* w_mlp_in  = (const float*)d_in[10];
    const float* b_mlp_in  = (const float*)d_in[11];
    const float* w_mlp_out = (const float*)d_in[12];
    const float* b_mlp_out = (const float*)d_in[13];
    float* out = (float*)d_out;

    // Workspace partition (floats).  Total ~1.79M floats (~7.2 MB).
    float* ws     = (float*)d_ws;
    float* vec    = ws;                               // 512*428
    float* gatesF = vec    + (long)SEQ * VEC_LD;      // 512*400
    float* gatesB = gatesF + (long)SEQ * G4H;         // 512*400
    float* cat    = gatesB + (long)SEQ * G4H;         // 512*628
    float* fbuf   = cat    + (long)SEQ * CAT_LD;      // 512*400
    float* gbuf   = fbuf   + (long)SEQ * D_MLP;       // 512*48
    float* wfP    = gbuf   + (long)SEQ * G_LD;        // 400*428
    float* wbP    = wfP    + (long)G4H * VEC_LD;      // 400*428
    float* wmP    = wbP    + (long)G4H * VEC_LD;      // 400*628
    float* woP    = wmP    + (long)D_MLP * CAT_LD;    // 48*400

    // 0) pack weights zero-padded (removes guards from GEMM hot loops)
    pack_w_kernel<<<G4H,    128, 0, stream>>>(w_ih_f,    G4H,   D_IN,  wfP, VEC_LD);
    pack_w_kernel<<<G4H,    128, 0, stream>>>(w_ih_b,    G4H,   D_IN,  wbP, VEC_LD);
    pack_w_kernel<<<D_MLP,  128, 0, stream>>>(w_mlp_in,  D_MLP, CAT_K, wmP, CAT_LD);
    pack_w_kernel<<<NREL_P, 128, 0, stream>>>(w_mlp_out, N_REL, D_MLP, woP, D_MLP);

    // 1) embedding gather
    gather_vec_kernel<<<SEQ, 128, 0, stream>>>(word_idx, pos_idx, ext_idx,
                                               word_emb, pos_emb, ext_emb, vec);

    // 2) LSTM gate GEMMs: [512,400] = vec[512,425] x Wih^T (+b)
    //    tiles = (512/16)*(400/16) = 800 -> 200 blocks of 4 wave32s
    wmma_gemm_f32<<<200, 128, 0, stream>>>(vec, wfP, VEC_LD,
                                           b_f, 1.0f, G4H,
                                           gatesF, G4H, G4H / 16);
    wmma_gemm_f32<<<200, 128, 0, stream>>>(vec, wbP, VEC_LD,
                                           b_b, 1.0f, G4H,
                                           gatesB, G4H, G4H / 16);

    // 3) LSTM activations + concat assembly
    lstm_cat_kernel<<<SEQ, 256, 0, stream>>>(gatesF, gatesB, vec, cat);

    // 4) f = cat[512,625] x w_mlp_in^T + b_mlp_in -> [512,400]
    wmma_gemm_f32<<<200, 128, 0, stream>>>(cat, wmP, CAT_LD,
                                           b_mlp_in, 1.0f, D_MLP,
                                           fbuf, D_MLP, D_MLP / 16);

    // 5) g = f[512,400] x w_mlp_out^T + 0.5*b_mlp_out -> [512,48] (42 valid)
    //    tiles = 32*3 = 96 -> 24 blocks of 4 wave32s
    wmma_gemm_f32<<<24, 128, 0, stream>>>(fbuf, woP, D_MLP,
                                          b_mlp_out, 0.5f, N_REL,
                                          gbuf, G_LD, G_LD / 16);

    // 6) scores = tanh(g_i + g_j) -> [512,512,42]
    scores_kernel<<<SEQ, 256, 0, stream>>>(gbuf, out);
}